// full_NN_40973988004616
// MI455X (gfx1250) — compile-verified
//
#include <hip/hip_runtime.h>
#include <cmath>

// Problem constants (match reference)
#define Bq 8192
#define Tq 64
#define Dq 100
#define Hq 110
#define Sq 63

constexpr float SIGc   = 0.3f;
constexpr float Rc     = 0.1f;
constexpr float DCOEFc = 0.1f;
constexpr float DTc    = 1.0f / 64.0f;
constexpr float EPSc   = 1e-6f;

typedef __attribute__((ext_vector_type(2))) float v2f;
typedef __attribute__((ext_vector_type(8))) float v8f;

#define MB   64     // batch rows per block
#define HS   116    // LDS row stride: 464B -> 16B aligned, conflict-free 16-row frag reads
#define NT   7      // 7 * 16 = 112 output columns
#define NPAD 112    // padded feature count / activation stride

// ---------------------------------------------------------------------------
// Pack weights into WMMA B-fragment order, zero-padded:
// dst[step][kcIdx][n(0..111)][kq(0..3)] = (k<K && n<N) ? w[step][kcIdx*4+kq][n] : 0
// Lane then loads one unconditional b64 (two k-values) at a constant offset.
// ---------------------------------------------------------------------------
__global__ void pack_w(const float* __restrict__ src, float* __restrict__ dst,
                       int K, int N, int NKC) {
  size_t per = (size_t)NKC * NPAD * 4;
  size_t tot = (size_t)Sq * per;
  size_t i = (size_t)blockIdx.x * blockDim.x + threadIdx.x;
  if (i >= tot) return;
  int s      = (int)(i / per);
  size_t rem = i - (size_t)s * per;
  int kcIdx  = (int)(rem / (NPAD * 4));
  int rem2   = (int)(rem % (NPAD * 4));
  int n  = rem2 >> 2;
  int kq = rem2 & 3;
  int k  = kcIdx * 4 + kq;
  float v = 0.0f;
  if (k < K && n < N) v = src[((size_t)s * K + k) * N + n];
  dst[i] = v;
}

__global__ void pack_b(const float* __restrict__ src, float* __restrict__ dst) {
  int i = blockIdx.x * blockDim.x + threadIdx.x;
  if (i >= Sq * NPAD) return;
  int s = i / NPAD, n = i - s * NPAD;
  dst[i] = (n < Dq) ? src[(size_t)s * Dq + n] : 0.0f;
}

// ---------------------------------------------------------------------------
// init: hx[:,0:100] = pi/2, hx[:,100] = y_init, hx[:,101:112] = 0; zero stats0
// ---------------------------------------------------------------------------
__global__ void init_k(float* __restrict__ hx, float* __restrict__ stats0,
                       const float* __restrict__ y_init) {
  int i = blockIdx.x * blockDim.x + threadIdx.x;
  if (i < Bq * NPAD) {
    int c = i % NPAD;
    float v = 0.0f;
    if (c < Dq)       v = 1.57079632679f;
    else if (c == Dq) v = y_init[0];
    hx[i] = v;
  }
  if (i < 256) stats0[i] = 0.0f;
}

// ---------------------------------------------------------------------------
// state update (one thread per batch row):
//   x += SIG*dw*y ; y -= drift*dt - sum(z*dw) ; z = bn3(u3)/D (or z_init)
// Accumulates BN0 stats of h=[x,y] (101 features). hx stride NPAD, y at col 100.
// ---------------------------------------------------------------------------
__global__ void state_k(const float* __restrict__ W, int t,
                        float* __restrict__ hx,
                        const float* __restrict__ u3,        // stride NPAD
                        const float* __restrict__ stats3,    // sum[128], sq[128]
                        const float* __restrict__ g3p, const float* __restrict__ b3p,
                        const float* __restrict__ z_init, int use_zinit,
                        float ec,
                        float* __restrict__ s0sum, float* __restrict__ s0sq,
                        float* __restrict__ statz,
                        float* __restrict__ out)             // may be null
{
  __shared__ float sc3[Dq], sh3[Dq];
  __shared__ float ssum[Dq + 1], ssq[Dq + 1];
  int tid = threadIdx.x;
  if (tid < Dq + 1) { ssum[tid] = 0.0f; ssq[tid] = 0.0f; }
  if (!use_zinit && tid < Dq) {
    float m  = stats3[tid] * (1.0f / Bq);
    float vv = stats3[128 + tid] * (1.0f / Bq) - m * m;
    float sc = g3p[tid] * rsqrtf(vv + EPSc);
    sc3[tid] = sc;
    sh3[tid] = b3p[tid] - m * sc;
  }
  if (blockIdx.x == 0) { statz[tid] = 0.0f; statz[tid + 128] = 0.0f; }
  __syncthreads();

  int r = blockIdx.x * blockDim.x + tid;
  float* hr = hx + (size_t)r * NPAD;
  const float* ur = u3 + (size_t)r * NPAD;
  float yo = hr[Dq];
  float zi = use_zinit ? z_init[0] : 0.0f;
  const float* Wr0 = W + ((size_t)r * Tq + t) * Dq;
  const float* Wr1 = Wr0 + Dq;

  float s = 0.0f, zs = 0.0f;
  for (int k = 0; k < Dq; ++k) {
    float dw = Wr1[k] - Wr0[k];
    float xk = hr[k] + SIGc * dw * yo;
    hr[k] = xk;
    if (out) out[(size_t)r * Dq + k] = xk;
    s += sinf(xk);
    float zk = use_zinit ? zi : (sc3[k] * ur[k] + sh3[k]) * (1.0f / Dq);
    zs += zk * dw;
    atomicAdd(&ssum[k], xk); atomicAdd(&ssq[k], xk * xk);
  }
  float ds3   = DCOEFc * s;
  float drift = -Rc * yo + 0.5f * SIGc * SIGc * ec * ds3 * ds3 * ds3;
  float yn = yo - drift * DTc + zs;
  hr[Dq] = yn;
  if (out) out[(size_t)Bq * Dq + r] = yn;
  atomicAdd(&ssum[Dq], yn); atomicAdd(&ssq[Dq], yn * yn);

  __syncthreads();
  if (tid < Dq + 1) {
    atomicAdd(&s0sum[tid], ssum[tid]);
    atomicAdd(&s0sq[tid],  ssq[tid]);
  }
}

// ---------------------------------------------------------------------------
// Fused BN(+ReLU) -> f32 WMMA GEMM -> (+bias) -> store U (stride 112) ->
// output BN stats. Block = 128 threads (4 wave32), 64 rows/block.
// Inner loop is branch-free: A from LDS (ds_load_b64), B from packed global
// (unconditional global_load_b64 at immediate offsets).
// ---------------------------------------------------------------------------
__global__ void gemm_k(const float* __restrict__ Hin, int K,
                       const float* __restrict__ wpk,     // [NKC][112][4]
                       const float* __restrict__ biasp,   // [112] padded or null
                       const float* __restrict__ sin_sum, const float* __restrict__ sin_sq,
                       const float* __restrict__ gin, const float* __restrict__ bin,
                       int relu_in,
                       float* __restrict__ Uout,          // stride NPAD
                       float* __restrict__ sout_sum, float* __restrict__ sout_sq,
                       float* __restrict__ statz)
{
  __shared__ float hbuf[MB * HS];
  __shared__ float scl[NPAD], sft[NPAD];
  __shared__ float osum[NPAD], osq[NPAD];
  int tid = threadIdx.x;

  if (tid < NPAD) {
    float sc = 0.0f, sh = 0.0f;
    if (tid < K) {
      float m  = sin_sum[tid] * (1.0f / Bq);
      float vv = sin_sq[tid] * (1.0f / Bq) - m * m;
      sc = gin[tid] * rsqrtf(vv + EPSc);
      sh = bin[tid] - m * sc;
    }
    scl[tid] = sc; sft[tid] = sh;
    osum[tid] = 0.0f; osq[tid] = 0.0f;
  }
  if (blockIdx.x == 0) { statz[tid] = 0.0f; statz[tid + 128] = 0.0f; }
  __syncthreads();

  int row0 = blockIdx.x * MB;
  // stage 64 x 112 normalized tile: float4 global loads, float4 LDS stores
  for (int i = tid; i < MB * (NPAD / 4); i += 128) {
    int rr = i / (NPAD / 4);
    int c4 = (i - rr * (NPAD / 4)) * 4;
    float4 v = *(const float4*)&Hin[(size_t)(row0 + rr) * NPAD + c4];
    float h0 = scl[c4 + 0] * v.x + sft[c4 + 0];
    float h1 = scl[c4 + 1] * v.y + sft[c4 + 1];
    float h2 = scl[c4 + 2] * v.z + sft[c4 + 2];
    float h3 = scl[c4 + 3] * v.w + sft[c4 + 3];
    if (relu_in) {
      h0 = fmaxf(h0, 0.0f); h1 = fmaxf(h1, 0.0f);
      h2 = fmaxf(h2, 0.0f); h3 = fmaxf(h3, 0.0f);
    }
    *(float4*)&hbuf[rr * HS + c4] = make_float4(h0, h1, h2, h3);
  }
  __syncthreads();

  int wave = tid >> 5, lane = tid & 31;
  int m0  = wave * 16;
  int lml = lane & 15;
  int hi  = (lane < 16) ? 0 : 2;            // ISA 16x4 A-layout k sub-offset
  v8f acc[NT] = {};

  int nkc = ((K + 3) & ~3) >> 2;
  const v2f* wl = (const v2f*)wpk + (size_t)lml * 2 + ((lane < 16) ? 0 : 1);
  int aoff = (m0 + lml) * HS + hi;
  for (int kc = 0; kc < nkc; ++kc) {
    v2f a = *(const v2f*)&hbuf[aoff];
    aoff += 4;
#pragma unroll
    for (int nt = 0; nt < NT; ++nt) {
      acc[nt] = __builtin_amdgcn_wmma_f32_16x16x4_f32(
          false, a, false, wl[nt * 32], (short)0, acc[nt], false, false);
    }
    wl += 2 * NPAD;
  }

  // epilogue: bias, store all 112 cols, per-feature stats
  int rbase = row0 + m0 + ((lane < 16) ? 0 : 8);
#pragma unroll
  for (int nt = 0; nt < NT; ++nt) {
    int n = nt * 16 + lml;
    float bv = biasp ? biasp[n] : 0.0f;
    float psum = 0.0f, psq = 0.0f;
#pragma unroll
    for (int j = 0; j < 8; ++j) {
      float v = acc[nt][j] + bv;
      Uout[(size_t)(rbase + j) * NPAD + n] = v;
      psum += v; psq += v * v;
    }
    atomicAdd(&osum[n], psum);
    atomicAdd(&osq[n], psq);
  }
  __syncthreads();
  if (tid < NPAD) {
    atomicAdd(&sout_sum[tid], osum[tid]);
    atomicAdd(&sout_sq[tid],  osq[tid]);
  }
}

// ---------------------------------------------------------------------------
// Driver
// ---------------------------------------------------------------------------
extern "C" void kernel_launch(void* const* d_in, const int* in_sizes, int n_in,
                              void* d_out, int out_size, void* d_ws, size_t ws_size,
                              hipStream_t stream)
{
  const float* W      = (const float*)d_in[0];
  const float* y_init = (const float*)d_in[1];
  const float* z_init = (const float*)d_in[2];
  const float* g0 = (const float*)d_in[3];
  const float* b0 = (const float*)d_in[4];
  const float* g1 = (const float*)d_in[5];
  const float* b1 = (const float*)d_in[6];
  const float* g2 = (const float*)d_in[7];
  const float* b2 = (const float*)d_in[8];
  const float* g3 = (const float*)d_in[9];
  const float* b3 = (const float*)d_in[10];
  const float* w0 = (const float*)d_in[11];
  const float* w1 = (const float*)d_in[12];
  const float* w2 = (const float*)d_in[13];
  const float* bias2 = (const float*)d_in[14];

  const int NKC0 = 26, NKC1 = 28, NKC2 = 28;
  const size_t P0 = (size_t)NKC0 * NPAD * 4;   // 11648 floats per step
  const size_t P1 = (size_t)NKC1 * NPAD * 4;   // 12544
  const size_t P2 = (size_t)NKC2 * NPAD * 4;   // 12544

  float* ws = (float*)d_ws;                    // ~24 MB used
  float* hx = ws;                              // B x 112 (y folded at col 100)
  float* u1 = hx + (size_t)Bq * NPAD;
  float* u2 = u1 + (size_t)Bq * NPAD;
  float* u3 = u2 + (size_t)Bq * NPAD;
  float* st = u3 + (size_t)Bq * NPAD;
  float* s0 = st;                              // each: sum[128] + sq[128]
  float* s1 = st + 256;
  float* s2 = st + 512;
  float* s3 = st + 768;
  float* dmy = st + 1024;                      // 512 floats dummy stats sink
  float* wp0 = dmy + 512;
  float* wp1 = wp0 + Sq * P0;
  float* wp2 = wp1 + Sq * P1;
  float* bp2 = wp2 + Sq * P2;

  // Pre-pack weights (graph-friendly, deterministic, runs every launch)
  pack_w<<<(int)((Sq * P0 + 255) / 256), 256, 0, stream>>>(w0, wp0, Dq + 1, Hq, NKC0);
  pack_w<<<(int)((Sq * P1 + 255) / 256), 256, 0, stream>>>(w1, wp1, Hq, Hq, NKC1);
  pack_w<<<(int)((Sq * P2 + 255) / 256), 256, 0, stream>>>(w2, wp2, Hq, Dq, NKC2);
  pack_b<<<(Sq * NPAD + 255) / 256, 256, 0, stream>>>(bias2, bp2);

  init_k<<<(Bq * NPAD + 255) / 256, 256, 0, stream>>>(hx, s0, y_init);

  for (int t = 0; t < Sq; ++t) {
    float ec = expf(-3.0f * Rc * (1.0f - t * DTc));
    state_k<<<Bq / 128, 128, 0, stream>>>(
        W, t, hx, u3, s3,
        g3 + (size_t)(t > 0 ? t - 1 : 0) * Dq,
        b3 + (size_t)(t > 0 ? t - 1 : 0) * Dq,
        z_init, (t == 0) ? 1 : 0, ec,
        s0, s0 + 128, s1, nullptr);
    gemm_k<<<Bq / MB, 128, 0, stream>>>(
        hx, Dq + 1, wp0 + (size_t)t * P0, nullptr,
        s0, s0 + 128, g0 + (size_t)t * (Dq + 1), b0 + (size_t)t * (Dq + 1), 0,
        u1, s1, s1 + 128, s2);
    gemm_k<<<Bq / MB, 128, 0, stream>>>(
        u1, Hq, wp1 + (size_t)t * P1, nullptr,
        s1, s1 + 128, g1 + (size_t)t * Hq, b1 + (size_t)t * Hq, 1,
        u2, s2, s2 + 128, s3);
    gemm_k<<<Bq / MB, 128, 0, stream>>>(
        u2, Hq, wp2 + (size_t)t * P2, bp2 + (size_t)t * NPAD,
        s2, s2 + 128, g2 + (size_t)t * Hq, b2 + (size_t)t * Hq, 1,
        u3, s3, s3 + 128, s0);
  }

  // Final Euler step (reference reuses dW[S-1]); writes (x, y) to d_out.
  float ecL = expf(-3.0f * Rc * (1.0f - (Sq - 1) * DTc));
  state_k<<<Bq / 128, 128, 0, stream>>>(
      W, Sq - 1, hx, u3, s3,
      g3 + (size_t)(Sq - 1) * Dq, b3 + (size_t)(Sq - 1) * Dq,
      z_init, 0, ecL,
      dmy, dmy + 128, dmy + 256, (float*)d_out);
}